// YOLOv7_Plus_21517786153048
// MI455X (gfx1250) — compile-verified
//
#include <hip/hip_runtime.h>
#include <hip/hip_bf16.h>
#include <stdint.h>

// ---------------- constants ----------------
#define NUM_CLASSES 80
#define TOPK        1000
#define NCAND       3000          // 3 * TOPK
#define SORTN       4096          // global sort size (pow2 >= NCAND)
#define CAP         8192          // per-level candidate cap (pow2), 64KB LDS sort
#define NBUCKET     2048          // 11-bit order-preserving float buckets
#define NTILE       188           // ceil(3000/16)
#define CONF_T      0.05f
#define NMS_T       0.6f

// ---------------- workspace layout (bytes, 16B aligned) ----------------
#define WS_HIST     0             // u32[3][2048]            24576
#define WS_CNT      24576         // u32[3] (pad16)
#define WS_THR      24592         // u32[3] (pad16)
#define WS_BMAX     24608         // u32[1] flipped-float max (pad16)
#define WS_PAIRS    24624         // u64[3*8192]             196608
#define WS_SCORES   221232        // f32[3000]
#define WS_LABELS   233232        // i32[3000]
#define WS_BOXES    245232        // f32[3000*4]
#define WS_SSCORE   293232        // f32[3000] sorted
#define WS_SLABEL   305232        // i32[3000] sorted
#define WS_KEEP     317232        // u32[188] keep-init bitmask (pad)
#define WS_SUP      317984        // u16[3000*188] suppress matrix 1128000
#define WS_TOTAL    1445984

__device__ __constant__ float d_anch[3][3][2] = {
    {{10.f, 13.f}, {16.f, 30.f}, {33.f, 23.f}},
    {{30.f, 61.f}, {62.f, 45.f}, {59.f, 119.f}},
    {{116.f, 90.f}, {156.f, 198.f}, {373.f, 326.f}}};

// order-preserving float <-> u32 (monotonic increasing)
__device__ __forceinline__ uint32_t flipf(float x) {
  uint32_t u = __float_as_uint(x);
  return u ^ (uint32_t)(((int32_t)u >> 31) | 0x80000000);
}
__device__ __forceinline__ float unflipf(uint32_t u) {
  u ^= (uint32_t)(((int32_t)(~u) >> 31) | 0x80000000);
  return __uint_as_float(u);
}

// ---------------- CDNA5 Tensor Data Mover: 1-D contiguous global -> LDS ----------------
typedef __attribute__((ext_vector_type(4))) uint32_t tdm_v4u;
typedef __attribute__((ext_vector_type(8))) uint32_t tdm_v8u;

// nelem must fit in 16 bits; dsz_code: 2 = 4-byte elements, 3 = 8-byte elements.
__device__ __forceinline__ void tdm_load_1d(uint32_t lds_off, const void* g,
                                            uint32_t nelem, uint32_t dsz_code) {
  uint64_t ga = (uint64_t)(uintptr_t)g;
  tdm_v4u g0;
  g0[0] = 1u;                                              // D# count=1, user mode
  g0[1] = lds_off;                                         // LDS byte address
  g0[2] = (uint32_t)ga;                                    // global_addr[31:0]
  g0[3] = (uint32_t)((ga >> 32) & 0x1FFFFFFu) | (2u << 30);// addr[56:32], type=2
  tdm_v8u g1;
  g1[0] = dsz_code << 16;                                  // data_size, wg_mask=0
  g1[1] = (nelem & 0xFFFFu) << 16;                         // tensor_dim0[15:0]
  g1[2] = ((nelem >> 16) & 0xFFFFu) | (1u << 16);          // tensor_dim0 hi, tensor_dim1=1
  g1[3] = (nelem & 0xFFFFu) << 16;                         // tile_dim0 = nelem
  g1[4] = 0u;                                              // tile_dim1/2 unused
  g1[5] = nelem;                                           // tensor_dim0_stride lo
  g1[6] = 0u;
  g1[7] = 0u;
  asm volatile("tensor_load_to_lds %0, %1" :: "s"(g0), "s"(g1) : "memory");
}

// ---------------- LDS bitonic sort, descending on u64 keys ----------------
__device__ __forceinline__ void bitonic_desc(unsigned long long* s, int N) {
  for (int k = 2; k <= N; k <<= 1) {
    for (int j = k >> 1; j > 0; j >>= 1) {
      __syncthreads();
      for (int i = threadIdx.x; i < N; i += blockDim.x) {
        int ixj = i ^ j;
        if (ixj > i) {
          unsigned long long a = s[i], b = s[ixj];
          bool sw = ((i & k) != 0) ? (a > b) : (a < b);
          if (sw) { s[i] = b; s[ixj] = a; }
        }
      }
    }
  }
  __syncthreads();
}

// ---------------- kernels ----------------
__global__ __launch_bounds__(256) void k_zero(uint32_t* __restrict__ p, int n) {
  for (int i = blockIdx.x * blockDim.x + threadIdx.x; i < n; i += gridDim.x * blockDim.x)
    p[i] = 0u;
}

__global__ __launch_bounds__(1024) void k_hist(const float4* __restrict__ cls4, int n4,
                                               uint32_t* __restrict__ gh) {
  __shared__ uint32_t h[NBUCKET];
  for (int i = threadIdx.x; i < NBUCKET; i += blockDim.x) h[i] = 0u;
  __syncthreads();
  for (int i = blockIdx.x * blockDim.x + threadIdx.x; i < n4; i += gridDim.x * blockDim.x) {
    float4 v = cls4[i];
    atomicAdd(&h[flipf(v.x) >> 21], 1u);
    atomicAdd(&h[flipf(v.y) >> 21], 1u);
    atomicAdd(&h[flipf(v.z) >> 21], 1u);
    atomicAdd(&h[flipf(v.w) >> 21], 1u);
  }
  __syncthreads();
  for (int i = threadIdx.x; i < NBUCKET; i += blockDim.x) {
    uint32_t c = h[i];
    if (c) atomicAdd(&gh[i], c);
  }
}

__global__ void k_thresh(const uint32_t* __restrict__ hist, uint32_t* __restrict__ thr) {
  if (threadIdx.x == 0) {
    const uint32_t* h = hist + blockIdx.x * NBUCKET;
    uint32_t acc = 0, t = 0;
    for (int b = NBUCKET - 1; b >= 0; --b) {
      acc += h[b];
      if (acc >= TOPK) { t = (uint32_t)b; break; }
    }
    thr[blockIdx.x] = t;
  }
}

__global__ __launch_bounds__(256) void k_compact(const float* __restrict__ cls, int n, int HW,
                                                 const uint32_t* __restrict__ thr,
                                                 uint32_t* __restrict__ cnt,
                                                 unsigned long long* __restrict__ pairs) {
  uint32_t t = *thr;
  for (int i = blockIdx.x * blockDim.x + threadIdx.x; i < n; i += gridDim.x * blockDim.x) {
    uint32_t key = flipf(cls[i]);
    if ((key >> 21) >= t) {
      uint32_t pos = atomicAdd(cnt, 1u);
      if (pos < CAP) {
        // reference flat index: ((p*3 + a)*80 + c) == p*240 + ch, ch = a*80+c
        uint32_t idx = (uint32_t)(i % HW) * 240u + (uint32_t)(i / HW);
        pairs[pos] = ((unsigned long long)key << 32) | (uint32_t)(~idx); // tie: idx asc
      }
    }
  }
}

__global__ __launch_bounds__(1024) void k_sortlevel(const float* __restrict__ regp, int W, int Hh,
                                                    float stridef, int lvl,
                                                    const unsigned long long* __restrict__ pairs_g,
                                                    float* __restrict__ s_out,
                                                    int* __restrict__ l_out,
                                                    float* __restrict__ b_out) {
  __shared__ unsigned long long sp[CAP];   // 64KB of the 320KB WGP LDS
  if (threadIdx.x < 32) {                  // wave 0 issues the TDM DMA
    tdm_load_1d((uint32_t)(uintptr_t)&sp[0], pairs_g, CAP, 3u);
    __builtin_amdgcn_s_wait_tensorcnt(0);
  }
  __syncthreads();
  bitonic_desc(sp, CAP);
  int r = threadIdx.x;
  if (r < TOPK) {
    unsigned long long pkt = sp[r];
    uint32_t key = (uint32_t)(pkt >> 32);
    uint32_t idx = ~(uint32_t)pkt;
    float logit = unflipf(key);
    float score = 1.f / (1.f + expf(-logit));
    uint32_t c = idx % 80u;
    uint32_t anc = idx / 80u;
    uint32_t a = anc % 3u;
    uint32_t p = anc / 3u;
    int HW = W * Hh;
    float x = (float)(p % (uint32_t)W);
    float y = (float)(p / (uint32_t)W);
    float tx = regp[(a * 4 + 0) * HW + p];
    float ty = regp[(a * 4 + 1) * HW + p];
    float tw = regp[(a * 4 + 2) * HW + p];
    float th = regp[(a * 4 + 3) * HW + p];
    float cx = (x + 0.5f) * stridef + tx * stridef;
    float cy = (y + 0.5f) * stridef + ty * stridef;
    float bw = expf(tw) * d_anch[lvl][a][0];
    float bh = expf(th) * d_anch[lvl][a][1];
    s_out[r] = score;
    l_out[r] = (int)c;
    b_out[4 * r + 0] = cx - 0.5f * bw;
    b_out[4 * r + 1] = cy - 0.5f * bh;
    b_out[4 * r + 2] = cx + 0.5f * bw;
    b_out[4 * r + 3] = cy + 0.5f * bh;
  }
}

__global__ __launch_bounds__(1024) void k_globalsort(const float* __restrict__ scores_all,
                                                     const int* __restrict__ labels_all,
                                                     const float* __restrict__ boxes_all,
                                                     float* __restrict__ out_boxes,
                                                     float* __restrict__ sscore,
                                                     int* __restrict__ slabel,
                                                     uint32_t* __restrict__ keep0,
                                                     uint32_t* __restrict__ bmax) {
  __shared__ unsigned long long sp[SORTN];
  __shared__ float ssc[NCAND];
  if (threadIdx.x < 32) {
    tdm_load_1d((uint32_t)(uintptr_t)&ssc[0], scores_all, NCAND, 2u);
    __builtin_amdgcn_s_wait_tensorcnt(0);
  }
  __syncthreads();
  for (int i = threadIdx.x; i < SORTN; i += blockDim.x)
    sp[i] = (i < NCAND)
                ? (((unsigned long long)__float_as_uint(ssc[i]) << 32) | (uint32_t)(~(uint32_t)i))
                : 0ull;  // scores in (0,1): positive float bits are monotone
  bitonic_desc(sp, SORTN);
  uint32_t lkey = 0u;
  for (int r = threadIdx.x; r < NCAND; r += blockDim.x) {
    unsigned long long pkt = sp[r];
    uint32_t orig = ~(uint32_t)pkt;
    float s = __uint_as_float((uint32_t)(pkt >> 32));
    sscore[r] = s;
    slabel[r] = labels_all[orig];
#pragma unroll
    for (int kk = 0; kk < 4; kk++) {
      float bv = boxes_all[4 * orig + kk];
      out_boxes[4 * r + kk] = bv;
      uint32_t fk = flipf(bv);
      if (fk > lkey) lkey = fk;
    }
  }
  atomicMax(bmax, lkey);  // global max(boxes) as order-preserving key
  __syncthreads();
  for (int t = threadIdx.x; t < NTILE; t += blockDim.x) {
    uint32_t m = 0;
    for (int b = 0; b < 16; b++) {
      int r = t * 16 + b;
      if (r < NCAND) {
        float s = __uint_as_float((uint32_t)(sp[r] >> 32));
        if (s > CONF_T) m |= (1u << b);
      }
    }
    keep0[t] = m;
  }
}

// 16x16 IoU tile per wave32; pairwise area-sum term via V_WMMA_F32_16X16X4_F32.
typedef __attribute__((ext_vector_type(2))) float v2f;
typedef __attribute__((ext_vector_type(8))) float v8f;

__global__ __launch_bounds__(32) void k_ioutile(const float* __restrict__ boxes,
                                                const int* __restrict__ labels,
                                                const uint32_t* __restrict__ bmax,
                                                unsigned short* __restrict__ sup) {
  int bi = blockIdx.y, bj = blockIdx.x;
  if (bj < bi) return;  // suppression needs j > i only
  int lane = threadIdx.x;
  int col = lane & 15;
  int half = lane >> 4;
  float om = unflipf(*bmax) + 1.0f;  // per-class offset scale
  __shared__ float ib[16][4];
  __shared__ float ia[16];
  if (lane < 16) {
    int i = bi * 16 + lane;
    float x1 = 0.f, y1 = 0.f, x2 = 0.f, y2 = 0.f;
    if (i < NCAND) {
      float o = (float)labels[i] * om;
      x1 = boxes[4 * i + 0] + o; y1 = boxes[4 * i + 1] + o;
      x2 = boxes[4 * i + 2] + o; y2 = boxes[4 * i + 3] + o;
    }
    ib[lane][0] = x1; ib[lane][1] = y1; ib[lane][2] = x2; ib[lane][3] = y2;
    ia[lane] = (x2 - x1) * (y2 - y1);
  }
  int j = bj * 16 + col;
  float jx1 = 0.f, jy1 = 0.f, jx2 = 0.f, jy2 = 0.f;
  if (j < NCAND) {
    float o = (float)labels[j] * om;
    jx1 = boxes[4 * j + 0] + o; jy1 = boxes[4 * j + 1] + o;
    jx2 = boxes[4 * j + 2] + o; jy2 = boxes[4 * j + 3] + o;
  }
  float ja = (jx2 - jx1) * (jy2 - jy1);
  __syncthreads();

  // D[m][n] = area_i[m] + area_j[n]:  A(16x4) = [area_i | 1 | 0 | 0], B(4x16) = [1 ; area_j ; 0 ; 0]
  // A layout: lanes 0-15 hold K=0,1; lanes 16-31 hold K=2,3 (zeros).
  v2f va, vb;
  va[0] = (half == 0) ? ia[lane & 15] : 0.f;
  va[1] = (half == 0) ? 1.f : 0.f;
  vb[0] = (half == 0) ? 1.f : 0.f;
  vb[1] = (half == 0) ? ja : 0.f;
  v8f cz = {0.f, 0.f, 0.f, 0.f, 0.f, 0.f, 0.f, 0.f};
  v8f dsum = __builtin_amdgcn_wmma_f32_16x16x4_f32(false, va, false, vb, (short)0, cz,
                                                   false, false);
#pragma unroll
  for (int k = 0; k < 8; k++) {
    int row = k + 8 * half;       // C/D layout: VGPR k = rows {k, k+8}
    int i = bi * 16 + row;
    float x1 = ib[row][0], y1 = ib[row][1], x2 = ib[row][2], y2 = ib[row][3];
    float iw = fminf(x2, jx2) - fmaxf(x1, jx1);
    float ih = fminf(y2, jy2) - fmaxf(y1, jy1);
    float inter = fmaxf(iw, 0.f) * fmaxf(ih, 0.f);
    float uni = dsum[k] - inter + 1e-10f;  // area_i + area_j - inter + eps
    bool pred = (i < NCAND) && (j < NCAND) && (j > i) && (inter > NMS_T * uni);
    unsigned long long bal = __ballot(pred ? 1 : 0);
    if (lane == 0) {
      int r0 = bi * 16 + k, r1 = r0 + 8;
      if (r0 < NCAND) sup[(size_t)r0 * NTILE + bj] = (unsigned short)(bal & 0xFFFFu);
      if (r1 < NCAND) sup[(size_t)r1 * NTILE + bj] = (unsigned short)((bal >> 16) & 0xFFFFu);
    }
  }
}

// sequential greedy scan (matches the reference lax.scan), single wave32
__global__ __launch_bounds__(32) void k_scan(const unsigned short* __restrict__ sup,
                                             const uint32_t* __restrict__ keep0,
                                             const float* __restrict__ sscore,
                                             const int* __restrict__ slabel,
                                             float* __restrict__ out) {
  __shared__ uint32_t kw[NTILE];
  int lane = threadIdx.x;
  for (int t = lane; t < NTILE; t += 32) kw[t] = keep0[t];
  __syncthreads();
  for (int i = 0; i < NCAND; i++) {
    uint32_t bit = (kw[i >> 4] >> (i & 15)) & 1u;  // uniform across the wave
    if (bit) {
      int bti = i >> 4;
      for (int t = lane; t < NTILE; t += 32)
        if (t >= bti) kw[t] &= ~(uint32_t)sup[(size_t)i * NTILE + t];
    }
    __syncthreads();
  }
  for (int r = lane; r < NCAND; r += 32) {
    uint32_t bit = (kw[r >> 4] >> (r & 15)) & 1u;
    out[12000 + r] = bit ? sscore[r] : 0.f;           // scores_out
    out[15000 + r] = bit ? (float)slabel[r] : -1.f;   // labels_out
    out[18000 + r] = bit ? 1.f : 0.f;                 // keep mask
  }
}

// ---------------- launcher ----------------
extern "C" void kernel_launch(void* const* d_in, const int* in_sizes, int n_in,
                              void* d_out, int out_size, void* d_ws, size_t ws_size,
                              hipStream_t stream) {
  const float* cls[3] = {(const float*)d_in[0], (const float*)d_in[2], (const float*)d_in[4]};
  const float* regp[3] = {(const float*)d_in[1], (const float*)d_in[3], (const float*)d_in[5]};
  static const int W[3] = {160, 80, 40};
  static const int H[3] = {160, 80, 40};
  static const float strd[3] = {8.f, 16.f, 32.f};

  char* ws = (char*)d_ws;
  uint32_t* hist = (uint32_t*)(ws + WS_HIST);
  uint32_t* cnt = (uint32_t*)(ws + WS_CNT);
  uint32_t* thr = (uint32_t*)(ws + WS_THR);
  uint32_t* bmax = (uint32_t*)(ws + WS_BMAX);
  unsigned long long* pairs = (unsigned long long*)(ws + WS_PAIRS);
  float* scores_all = (float*)(ws + WS_SCORES);
  int* labels_all = (int*)(ws + WS_LABELS);
  float* boxes_all = (float*)(ws + WS_BOXES);
  float* sscore = (float*)(ws + WS_SSCORE);
  int* slabel = (int*)(ws + WS_SLABEL);
  uint32_t* keep0 = (uint32_t*)(ws + WS_KEEP);
  unsigned short* sup = (unsigned short*)(ws + WS_SUP);
  float* out = (float*)d_out;

  k_zero<<<dim3(512), dim3(256), 0, stream>>>((uint32_t*)ws, WS_TOTAL / 4);

  for (int l = 0; l < 3; l++) {
    int n4 = 240 * W[l] * H[l] / 4;
    int g = (n4 + 1023) / 1024; if (g > 2048) g = 2048;
    k_hist<<<dim3(g), dim3(1024), 0, stream>>>((const float4*)cls[l], n4, hist + l * NBUCKET);
  }
  k_thresh<<<dim3(3), dim3(32), 0, stream>>>(hist, thr);
  for (int l = 0; l < 3; l++) {
    int n = 240 * W[l] * H[l];
    int g = (n + 255) / 256; if (g > 4096) g = 4096;
    k_compact<<<dim3(g), dim3(256), 0, stream>>>(cls[l], n, W[l] * H[l], thr + l, cnt + l,
                                                 pairs + (size_t)l * CAP);
  }
  for (int l = 0; l < 3; l++) {
    k_sortlevel<<<dim3(1), dim3(1024), 0, stream>>>(regp[l], W[l], H[l], strd[l], l,
                                                    pairs + (size_t)l * CAP,
                                                    scores_all + l * TOPK, labels_all + l * TOPK,
                                                    boxes_all + (size_t)l * TOPK * 4);
  }
  k_globalsort<<<dim3(1), dim3(1024), 0, stream>>>(scores_all, labels_all, boxes_all, out, sscore,
                                                   slabel, keep0, bmax);
  k_ioutile<<<dim3(NTILE, NTILE), dim3(32), 0, stream>>>(out, slabel, bmax, sup);
  k_scan<<<dim3(1), dim3(32), 0, stream>>>(sup, keep0, sscore, slabel, out);

  (void)in_sizes; (void)n_in; (void)out_size; (void)ws_size;
}